// BalancedSoftmaxCrossEntropy_91018946937483
// MI455X (gfx1250) — compile-verified
//
#include <hip/hip_runtime.h>
#include <hip/hip_bf16.h>
#include <math.h>

// ---------------------------------------------------------------------------
// Balanced softmax cross-entropy, MI455X (gfx1250, wave32).
// tau==2  =>  scaled weight w[t] = counts[t]^2 / sum(counts^2).
// Memory-bound: 524 MB logits read once (~22.5us @ 23.3TB/s).
// Each block stages one 32000-float row into LDS via the CDNA5 async
// global->LDS path; pass-1 (max) is software-pipelined against the in-flight
// async loads using in-order ASYNCcnt partial waits; pass-2 (sum exp) runs
// entirely from LDS. Single HBM read per logit element.
// ---------------------------------------------------------------------------

#define THREADS 512
#define MAX_C   32000   // row length; 125 KB LDS per block (320 KB/WGP -> 2 blocks)
#define NPART   64      // partial-sum blocks for the counts^2 reduction
#define KMAX    ((MAX_C / 4 + THREADS - 1) / THREADS)   // 16 staged chunks

#if defined(__gfx1250__) && \
    __has_builtin(__builtin_amdgcn_global_load_async_to_lds_b128) && \
    __has_builtin(__builtin_amdgcn_s_wait_asynccnt)
#define USE_ASYNC_LDS 1
#else
#define USE_ASYNC_LDS 0
#endif

#if USE_ASYNC_LDS
typedef int v4i_ __attribute__((vector_size(16)));
typedef __attribute__((address_space(1))) v4i_* g_v4i_ptr;
typedef __attribute__((address_space(3))) v4i_* l_v4i_ptr;
typedef __attribute__((address_space(1))) char* g_char_ptr;
typedef __attribute__((address_space(3))) char* l_char_ptr;
#endif

__device__ __forceinline__ float warp_max32(float v) {
#pragma unroll
  for (int o = 16; o > 0; o >>= 1) v = fmaxf(v, __shfl_xor(v, o, 32));
  return v;
}
__device__ __forceinline__ float warp_sum32(float v) {
#pragma unroll
  for (int o = 16; o > 0; o >>= 1) v += __shfl_xor(v, o, 32);
  return v;
}

#if USE_ASYNC_LDS
// Pass-1 step K: wait for staged chunk K (in-order ASYNCcnt), then fmax it.
template <int K>
struct Pass1Step {
  static __device__ __forceinline__ void run(const float4* t4, int C4, int tid,
                                             float& m) {
    __builtin_amdgcn_s_wait_asynccnt(KMAX - 1 - K);
    const int i = tid + K * THREADS;
    if (i < C4) {
      float4 v = t4[i];
      m = fmaxf(m, fmaxf(fmaxf(v.x, v.y), fmaxf(v.z, v.w)));
    }
    Pass1Step<K + 1>::run(t4, C4, tid, m);
  }
};
template <>
struct Pass1Step<KMAX> {
  static __device__ __forceinline__ void run(const float4*, int, int, float&) {}
};
#endif

// --- Kernel 1a: per-block partial sums of counts^2 (one load per thread) ----
__global__ __launch_bounds__(256)
void bsce_partial_sq(const float* __restrict__ counts, int C,
                     float* __restrict__ partial) {
  __shared__ float red[8];
  const int C4 = C >> 2;
  const float4* c4 = (const float4*)counts;
  float acc = 0.f;
  for (int i = blockIdx.x * 256 + threadIdx.x; i < C4; i += 256 * NPART) {
    float4 v = c4[i];
    acc += v.x * v.x + v.y * v.y + v.z * v.z + v.w * v.w;
  }
  if (blockIdx.x == 0) {  // scalar tail if C % 4 != 0
    for (int j = (C4 << 2) + threadIdx.x; j < C; j += 256) {
      float c = counts[j];
      acc += c * c;
    }
  }
  acc = warp_sum32(acc);
  const int lane = threadIdx.x & 31, w = threadIdx.x >> 5;
  if (lane == 0) red[w] = acc;
  __syncthreads();
  if (w == 0) {
    float v = (lane < 8) ? red[lane] : 0.f;
    v = warp_sum32(v);
    if (lane == 0) partial[blockIdx.x] = v;
  }
}

// --- Kernel 1b: s2 = sum of the 64 partials ---------------------------------
__global__ void bsce_s2(const float* __restrict__ partial,
                        float* __restrict__ s2_out) {
  __shared__ float red[2];
  float v = partial[threadIdx.x];          // launched with NPART == 64 threads
  v = warp_sum32(v);
  const int lane = threadIdx.x & 31, w = threadIdx.x >> 5;
  if (lane == 0) red[w] = v;
  __syncthreads();
  if (threadIdx.x == 0) s2_out[0] = red[0] + red[1];
}

// --- Kernel 2: per-row loss -------------------------------------------------
__global__ __launch_bounds__(THREADS)
void bsce_row(const float* __restrict__ logits, const int* __restrict__ targets,
              const float* __restrict__ counts, const float* __restrict__ s2p,
              float* __restrict__ rowloss, int C) {
  __shared__ __align__(16) float tile[MAX_C];
  __shared__ float red[THREADS / 32];

  const int row  = blockIdx.x;
  const float* src = logits + (size_t)row * (size_t)C;
  const int C4 = C >> 2;               // C==32000 -> 8000 exact float4s
  const int lane = threadIdx.x & 31, w = threadIdx.x >> 5;
  const int tid  = threadIdx.x;

  float m = -INFINITY;

#if USE_ASYNC_LDS
  // ---- Stage whole row into LDS (single HBM read per element). Every wave
  // issues a uniform KMAX async b128 loads (out-of-range lanes clamp to the
  // last element: duplicate same-value LDS writes, benign), so the in-order
  // per-wave ASYNCcnt lets pass-1 start on chunk k after waiting <= KMAX-1-k.
  {
    g_char_ptr g = (g_char_ptr)src;
    l_char_ptr l = (l_char_ptr)tile;
#pragma unroll
    for (int k = 0; k < KMAX; ++k) {
      int i = tid + k * THREADS;
      size_t j = (size_t)((i < C4) ? i : (C4 - 1));
      __builtin_amdgcn_global_load_async_to_lds_b128(
          (g_v4i_ptr)(g + 16 * j), (l_v4i_ptr)(l + 16 * j), 0, 0);
    }
  }
  // ---- Pass 1 (max), pipelined against the in-flight staging ----
  Pass1Step<0>::run((const float4*)tile, C4, tid, m);
#else
  for (int i = tid; i < C4; i += THREADS)
    ((float4*)tile)[i] = ((const float4*)src)[i];
  __syncthreads();
  for (int i = tid; i < C4; i += THREADS) {
    float4 v = ((const float4*)tile)[i];
    m = fmaxf(m, fmaxf(fmaxf(v.x, v.y), fmaxf(v.z, v.w)));
  }
#endif

  m = warp_max32(m);
  if (lane == 0) red[w] = m;
  __syncthreads();                     // also makes the full tile LDS-visible
  {
    float v = (lane < (THREADS / 32)) ? red[lane] : -INFINITY;
    m = warp_max32(v);                 // every wave computes identical m
  }
  __syncthreads();                     // red reused below

  // ---- Pass 2: Z = sum exp(x - m) (from LDS) ----
  float s = 0.f;
#pragma unroll 2
  for (int i = tid; i < C4; i += THREADS) {
    float4 v = ((const float4*)tile)[i];
    s += __expf(v.x - m) + __expf(v.y - m) + __expf(v.z - m) + __expf(v.w - m);
  }
  s = warp_sum32(s);
  if (lane == 0) red[w] = s;
  __syncthreads();

  if (tid == 0) {
    float Z = 0.f;
#pragma unroll
    for (int i = 0; i < THREADS / 32; ++i) Z += red[i];
    const int t   = targets[row];
    const float xt = tile[t];
    const float logp = xt - m - logf(Z);
    const float c = counts[t];
    const float wgt = (c * c) / s2p[0];
    rowloss[row] = -logp * wgt;
  }
}

// --- Kernel 3: deterministic mean over rows (one float4 per thread) ---------
__global__ __launch_bounds__(1024)
void bsce_final(const float* __restrict__ rowloss, int B,
                float* __restrict__ out) {
  __shared__ float red[32];
  const int B4 = B >> 2;               // B==4096 -> 1024 exact float4s
  const float4* r4 = (const float4*)rowloss;
  float acc = 0.f;
  for (int i = threadIdx.x; i < B4; i += 1024) {
    float4 v = r4[i];
    acc += v.x + v.y + v.z + v.w;
  }
  for (int j = (B4 << 2) + threadIdx.x; j < B; j += 1024)  // tail if B % 4 != 0
    acc += rowloss[j];
  acc = warp_sum32(acc);
  const int lane = threadIdx.x & 31, w = threadIdx.x >> 5;
  if (lane == 0) red[w] = acc;
  __syncthreads();
  if (w == 0) {
    float v = red[lane];               // blockDim 1024 -> exactly 32 waves
    v = warp_sum32(v);
    if (lane == 0) out[0] = v / (float)B;
  }
}

extern "C" void kernel_launch(void* const* d_in, const int* in_sizes, int n_in,
                              void* d_out, int out_size, void* d_ws, size_t ws_size,
                              hipStream_t stream) {
  const float* logits  = (const float*)d_in[0];
  const int*   targets = (const int*)d_in[1];
  const float* counts  = (const float*)d_in[2];
  const int B = in_sizes[1];       // 4096
  const int C = in_sizes[2];       // 32000

  float* wsf     = (float*)d_ws;
  float* s2      = wsf;            // 1 float @ offset 0
  float* partial = wsf + 64;       // NPART floats
  float* rowloss = wsf + 128;      // B floats
  float* out     = (float*)d_out;

  bsce_partial_sq<<<NPART, 256, 0, stream>>>(counts, C, partial);
  bsce_s2<<<1, NPART, 0, stream>>>(partial, s2);
  bsce_row<<<B, THREADS, 0, stream>>>(logits, targets, counts, s2, rowloss, C);
  bsce_final<<<1, 1024, 0, stream>>>(rowloss, B, out);
}